// Angles2Backbone_79585743995279
// MI455X (gfx1250) — compile-verified
//
#include <hip/hip_runtime.h>
#include <math.h>

typedef __attribute__((ext_vector_type(2))) float v2f;
typedef __attribute__((ext_vector_type(8))) float v8f;

#define LCONST 512
#define A3 (3*LCONST)      // 1536 transforms per row
#define NT 256             // threads per block (8 waves)
#define CH (A3/NT)         // 6 transforms per thread
#define OUTROW (3*A3)      // 4608 output floats per row

struct Aff {
  float r[9];   // 3x3 block, row major in registers
  float t[3];   // translation
};

__device__ __forceinline__ void affIdentity(Aff &a){
  a.r[0]=1.f;a.r[1]=0.f;a.r[2]=0.f;
  a.r[3]=0.f;a.r[4]=1.f;a.r[5]=0.f;
  a.r[6]=0.f;a.r[7]=0.f;a.r[8]=1.f;
  a.t[0]=0.f;a.t[1]=0.f;a.t[2]=0.f;
}

// A = A * B  (affine compose, bottom row implicitly [0,0,0,1])
__device__ __forceinline__ void affMulRight(Aff &A, const Aff &B){
  float nr[9], nt[3];
  #pragma unroll
  for (int i=0;i<3;i++){
    #pragma unroll
    for (int j=0;j<3;j++){
      nr[i*3+j] = A.r[i*3+0]*B.r[0*3+j] + A.r[i*3+1]*B.r[1*3+j] + A.r[i*3+2]*B.r[2*3+j];
    }
    nt[i] = A.r[i*3+0]*B.t[0] + A.r[i*3+1]*B.t[1] + A.r[i*3+2]*B.t[2] + A.t[i];
  }
  #pragma unroll
  for (int i=0;i<9;i++) A.r[i]=nr[i];
  A.t[0]=nt[0]; A.t[1]=nt[1]; A.t[2]=nt[2];
}

// Build transform j (= 3*residue + c) from angles; j==0 is identity.
__device__ __forceinline__ void makeB(int j,
                                      const float* __restrict__ phi,
                                      const float* __restrict__ psi,
                                      const float* __restrict__ omg,
                                      const float* ckv, const float* skv, const float* Rv,
                                      Aff &B){
  if (j==0){ affIdentity(B); return; }
  int r = j/3; int c = j - 3*r;
  float alpha;
  if (c==0)      alpha = (r==0) ? 0.f : psi[r-1];
  else if (c==1) alpha = (r==0) ? 0.f : omg[r-1];
  else           alpha = phi[r];
  float sa, ca;
  __sincosf(alpha, &sa, &ca);
  const float ck=ckv[c], sk=skv[c], R=Rv[c];
  B.r[0]=ck;     B.r[1]=sk;      B.r[2]=0.f;
  B.r[3]=sk*ca;  B.r[4]=-ck*ca;  B.r[5]=sa;
  B.r[6]=sk*sa;  B.r[7]=-ck*sa;  B.r[8]=-ca;
  B.t[0]=R*ck;   B.t[1]=R*sk*ca; B.t[2]=R*sk*sa;
}

// One V_WMMA_F32_16X16X4_F32 combining 4 independent 4x4 products:
//   for the 4 matrices of one group: dOut[tt] = (useI ? I : sA[aIdx]) * sB[bIdx]
// Matrices are stored COLUMN-MAJOR (16 floats, elem(row,col) at col*4+row) so
// the B operand is one ds_load_b64 and the diagonal-block scatter is a single
// ds_store_b128 from the 16 lanes that own diagonal data.
__device__ __forceinline__ void wmma4_step(const float* sA, int aIdx, bool useI,
                                           const float* sB, int bIdx,
                                           float* dOut, int oIdx,
                                           int rr, int kA, bool active, bool hiSel)
{
  v2f a;
  a.x = sA[aIdx*16 + kA*4 + rr];        // A[m][kA]   (column-major: col kA, row rr)
  a.y = sA[aIdx*16 + (kA+1)*4 + rr];    // A[m][kA+1]
  if (useI){ a.x = (rr==kA)   ? 1.f : 0.f;
             a.y = (rr==kA+1) ? 1.f : 0.f; }
  const v2f bv = *(const v2f*)&sB[bIdx*16 + rr*4 + kA];   // B[kA..kA+1][n] contiguous
  v8f c = {};
  c = __builtin_amdgcn_wmma_f32_16x16x4_f32(
          false, a, false, bv, (short)0, c, false, false);
  // Diagonal block elements this lane may own: C VGPRs [hiSel*4 .. +3]
  float4 val;
  val.x = hiSel ? c[4] : c[0];
  val.y = hiSel ? c[5] : c[1];
  val.z = hiSel ? c[6] : c[2];
  val.w = hiSel ? c[7] : c[3];
  if (active) *(float4*)&dOut[oIdx*16 + rr*4] = val;   // column rr, rows 0..3
}

__global__ __launch_bounds__(NT)
void a2b_scan_kernel(const float* __restrict__ angles,
                     const float* __restrict__ param,
                     const int*   __restrict__ alen,
                     float*       __restrict__ out)
{
  __shared__ float bufA[NT*16];    // 16 KB ping
  __shared__ float bufB[NT*16];    // 16 KB pong
  __shared__ float wmidA[8*16];    // wave totals ping
  __shared__ float wmidB[8*16];    // wave totals pong
  __shared__ float Ew[8*16];       // exclusive wave prefixes

  const int b = blockIdx.x;
  const int t = threadIdx.x;
  const float* ang = angles + (size_t)b * 3 * LCONST;
  const float* phi = ang;
  const float* psi = ang + LCONST;
  const float* omg = ang + 2*LCONST;

  // Per-c constants: kappa = [p5,p1,p3], Rlen = [p4,p0,p2]
  float ckv[3], skv[3], Rv[3];
  {
    const float p0=param[0], p1=param[1], p2=param[2],
                p3=param[3], p4=param[4], p5=param[5];
    __sincosf(p5, &skv[0], &ckv[0]);
    __sincosf(p1, &skv[1], &ckv[1]);
    __sincosf(p3, &skv[2], &ckv[2]);
    Rv[0]=p4; Rv[1]=p0; Rv[2]=p2;
  }

  // ---- Phase 0: per-thread chunk product; write column-major into bufA ----
  Aff cur; affIdentity(cur);
  const int j0 = t * CH;
  #pragma unroll
  for (int jj=0; jj<CH; jj++){
    Aff Bm; makeB(j0+jj, phi, psi, omg, ckv, skv, Rv, Bm);
    affMulRight(cur, Bm);
  }
  {
    float* d = &bufA[t*16];
    *(float4*)&d[0]  = make_float4(cur.r[0], cur.r[3], cur.r[6], 0.f);
    *(float4*)&d[4]  = make_float4(cur.r[1], cur.r[4], cur.r[7], 0.f);
    *(float4*)&d[8]  = make_float4(cur.r[2], cur.r[5], cur.r[8], 0.f);
    *(float4*)&d[12] = make_float4(cur.t[0], cur.t[1], cur.t[2], 1.f);
  }

  // Lane roles for WMMA packing
  const int lane = t & 31;
  const int wv   = t >> 5;
  const int half = lane >> 4;          // 0: lanes 0-15, 1: lanes 16-31
  const int m    = lane & 15;          // A row / B col
  const int sub  = m >> 2;             // which 4x4 block (0..3)
  const int rr   = m & 3;              // row/col within block
  const int kA   = half * 2;           // K pair for this lane half
  const bool active = ((sub>>1) == half);
  const bool hiSel  = (sub & 1) != 0;
  const int wbase = wv * 32;

  // ---- Phase 1a: intra-wave scan of 32 matrices, NO barriers needed ----
  // (each wave only touches its own 32-entry slice; same-wave LDS is ordered)
  {
    float* src = bufA; float* dst = bufB;
    #pragma unroll
    for (int si=0; si<5; si++){
      const int s = 1 << si;
      #pragma unroll
      for (int q=0; q<8; q++){
        const int tt = wbase + 4*q + sub;
        const int li = tt - s;
        const bool useI = (li < wbase);
        const int liS = useI ? tt : li;
        wmma4_step(src, liS, useI, src, tt, dst, tt, rr, kA, active, hiSel);
      }
      float* tmp = src; src = dst; dst = tmp;
    }
    // 5 steps -> intra-wave inclusive prefixes live in bufB
  }
  __syncthreads();   // barrier #1: wave totals visible to wave 0

  // ---- Phase 1b: wave 0 scans the 8 wave totals, emits exclusive prefixes ----
  if (wv == 0){
    {
      const int w8  = lane >> 2;   // 0..7
      const int prt = lane & 3;    // 0..3 (column group)
      *(float4*)&wmidA[w8*16 + prt*4] =
          *(const float4*)&bufB[(w8*32 + 31)*16 + prt*4];
    }
    float* msrc = wmidA; float* mdst = wmidB;
    #pragma unroll
    for (int si=0; si<3; si++){
      const int s = 1 << si;
      #pragma unroll
      for (int q=0; q<2; q++){
        const int tt = 4*q + sub;
        const int li = tt - s;
        const bool useI = (li < 0);
        const int liS = useI ? tt : li;
        wmma4_step(msrc, liS, useI, msrc, tt, mdst, tt, rr, kA, active, hiSel);
      }
      float* tmp = msrc; msrc = mdst; mdst = tmp;
    }
    // msrc now holds inclusive totals scan; write exclusive prefixes
    {
      const int w8  = lane >> 2;
      const int prt = lane & 3;
      float4 val;
      if (w8 == 0){
        val.x = (prt==0)?1.f:0.f;
        val.y = (prt==1)?1.f:0.f;
        val.z = (prt==2)?1.f:0.f;
        val.w = (prt==3)?1.f:0.f;
      } else {
        val = *(const float4*)&msrc[(w8-1)*16 + prt*4];
      }
      *(float4*)&Ew[w8*16 + prt*4] = val;
    }
  }
  __syncthreads();   // barrier #2: Ew visible to all waves

  // ---- Phase 1c: apply exclusive wave prefix in place: P[tt] = Ew[wv]*intra[tt] ----
  #pragma unroll
  for (int q=0; q<8; q++){
    const int tt = wbase + 4*q + sub;
    wmma4_step(Ew, wv, false, bufB, tt, bufB, tt, rr, kA, active, hiSel);
  }
  __syncthreads();   // barrier #3: full prefixes visible for phase 2

  // ---- Phase 2: per-thread exclusive prefix, re-walk chunk, emit coords ----
  Aff E;
  if (t == 0){
    affIdentity(E);
  } else {
    const float* sgl = &bufB[(t-1)*16];   // column-major
    E.r[0]=sgl[0];  E.r[3]=sgl[1];  E.r[6]=sgl[2];
    E.r[1]=sgl[4];  E.r[4]=sgl[5];  E.r[7]=sgl[6];
    E.r[2]=sgl[8];  E.r[5]=sgl[9];  E.r[8]=sgl[10];
    E.t[0]=sgl[12]; E.t[1]=sgl[13]; E.t[2]=sgl[14];
  }

  const int len3 = 3 * alen[b];
  float* orow = out + (size_t)b * OUTROW;
  #pragma unroll
  for (int jj=0; jj<CH; jj++){
    const int j = j0 + jj;
    Aff Bm; makeB(j, phi, psi, omg, ckv, skv, Rv, Bm);
    affMulRight(E, Bm);
    const float msk = (j < len3) ? 1.f : 0.f;
    orow[j*3+0] = E.t[0] * msk;
    orow[j*3+1] = E.t[1] * msk;
    orow[j*3+2] = E.t[2] * msk;
  }
}

extern "C" void kernel_launch(void* const* d_in, const int* in_sizes, int n_in,
                              void* d_out, int out_size, void* d_ws, size_t ws_size,
                              hipStream_t stream) {
  (void)n_in; (void)out_size; (void)d_ws; (void)ws_size;
  const float* angles = (const float*)d_in[0];   // (256, 3, 512) f32
  const float* param  = (const float*)d_in[1];   // (6,) f32
  const int*   alen   = (const int*)  d_in[2];   // (256,) i32
  float* out = (float*)d_out;                    // (256, 4608) f32
  const int batch = in_sizes[2];
  a2b_scan_kernel<<<dim3(batch), dim3(NT), 0, stream>>>(angles, param, alen, out);
}